// GCNLayer_83751862272172
// MI455X (gfx1250) — compile-verified
//
#include <hip/hip_runtime.h>

#define N_NODES 100000
#define D 64
#define EPS 1e-12f

typedef float v2f __attribute__((ext_vector_type(2)));
typedef float v8f __attribute__((ext_vector_type(8)));

// ---------------------------------------------------------------------------
// Scatter phase: one wave32 per edge. Lane l handles floats [2l, 2l+1] of the
// 64-wide message. Source row gather is one coalesced 256B burst per wave
// (global_load_b64 per lane); accumulation is global_atomic_add_f32 into the
// destination row. All of feat/out (~100 MB) is L2-resident on MI455X (192 MB
// L2), so atomics RMW in L2. Edge streams use non-temporal loads to avoid
// evicting the hot working set with 96 MB of single-use data.
// ---------------------------------------------------------------------------
__global__ void __launch_bounds__(256)
scatter_edges(const float* __restrict__ feat,
              const float* __restrict__ norm,
              const int*   __restrict__ src,
              const int*   __restrict__ dst,
              float*       __restrict__ out,
              int n_edges)
{
    int gid  = blockIdx.x * blockDim.x + threadIdx.x;
    int edge = gid >> 5;              // wave32: one edge per wave
    int lane = threadIdx.x & 31;
    if (edge >= n_edges) return;      // wave-uniform exit

    int   s = __builtin_nontemporal_load(src  + edge);
    int   d = __builtin_nontemporal_load(dst  + edge);
    float w = __builtin_nontemporal_load(norm + edge);

    const float2* __restrict__ srow = (const float2*)(feat + (size_t)s * D);
    float2 v = srow[lane];

    float* drow = out + (size_t)d * D + lane * 2;
    atomicAdd(drow + 0, v.x * w);
    atomicAdd(drow + 1, v.y * w);
}

// ---------------------------------------------------------------------------
// Normalize phase: one wave32 per 16-row tile of H (in place in d_out).
// Row sums-of-squares = diagonal of G = H_tile x H_tile^T, accumulated with
// 16x V_WMMA_F32_16X16X4_F32 over K=64 (full f32 precision).
// f32 A-operand 16x4 layout: lanes 0-15 -> {K0,K1}, lanes 16-31 -> {K2,K3},
// M = lane&15. The B-operand of A*A^T has identical per-lane contents, so the
// same register pair feeds SRC0 and SRC1.
// C/D layout: lanes 0-15, VGPR r -> G[r, lane]; lanes 16-31 -> G[r+8, lane-16].
// ---------------------------------------------------------------------------
__global__ void __launch_bounds__(256)
l2norm_rows(float* __restrict__ h, int n_rows)
{
    int wave = (blockIdx.x * blockDim.x + threadIdx.x) >> 5;
    int lane = threadIdx.x & 31;
    int row0 = wave * 16;
    if (row0 >= n_rows) return;       // wave-uniform: EXEC all-ones for WMMA

    const int m  = lane & 15;         // matrix row within tile
    const int kh = lane >> 4;         // K-half select
    float* __restrict__ base = h + (size_t)row0 * D;

    v8f acc = {};
    #pragma unroll
    for (int c = 0; c < 16; ++c) {    // K = 64 in chunks of 4
        int col = c * 4 + kh * 2;
        v2f a;
        a.x = base[(size_t)m * D + col + 0];
        a.y = base[(size_t)m * D + col + 1];
        // G += A * A^T   (16x16x4 f32 WMMA; B operand == A operand layout)
        acc = __builtin_amdgcn_wmma_f32_16x16x4_f32(
            false, a, false, a, (short)0, acc, false, false);
    }

    // Extract diagonal G[r,r] -> per-row inverse norms (broadcast to all lanes)
    float scale[16];
    #pragma unroll
    for (int r = 0; r < 8; ++r) {
        float dg = __shfl(acc[r], r, 32);           // G[r,r]: lane r, VGPR r
        scale[r] = 1.0f / fmaxf(sqrtf(dg), EPS);
    }
    #pragma unroll
    for (int r = 8; r < 16; ++r) {
        float dg = __shfl(acc[r - 8], r + 16, 32);  // G[r,r]: lane r+16, VGPR r-8
        scale[r] = 1.0f / fmaxf(sqrtf(dg), EPS);
    }

    // Scale rows in place: per row, 32 lanes x float2 = 64 floats.
    #pragma unroll
    for (int r = 0; r < 16; ++r) {
        float2* rp = (float2*)(base + (size_t)r * D);
        float2 v = rp[lane];
        v.x *= scale[r];
        v.y *= scale[r];
        rp[lane] = v;
    }
}

extern "C" void kernel_launch(void* const* d_in, const int* in_sizes, int n_in,
                              void* d_out, int out_size, void* d_ws, size_t ws_size,
                              hipStream_t stream)
{
    const float* user_feat = (const float*)d_in[0];
    const float* item_feat = (const float*)d_in[1];
    const float* norm_ui   = (const float*)d_in[2];
    const float* norm_iu   = (const float*)d_in[3];
    const int*   src_ui    = (const int*)  d_in[4];
    const int*   dst_ui    = (const int*)  d_in[5];
    const int*   src_iu    = (const int*)  d_in[6];
    const int*   dst_iu    = (const int*)  d_in[7];

    float* out = (float*)d_out;
    const int n_edges = in_sizes[4];

    // d_out is poisoned by the harness; zero the accumulators every call.
    hipMemsetAsync(out, 0, (size_t)out_size * sizeof(float), stream);

    float* user_h = out;                        // output slot 0: user embeddings
    float* item_h = out + (size_t)N_NODES * D;  // output slot 1: item embeddings

    // 8 edges (8 waves) per 256-thread block.
    int eblocks = (n_edges + 7) / 8;
    // user_h = segment_sum(norm_iu * item_feat[src_iu]) over dst_iu
    scatter_edges<<<eblocks, 256, 0, stream>>>(item_feat, norm_iu, src_iu, dst_iu,
                                               user_h, n_edges);
    // item_h = segment_sum(norm_ui * user_feat[src_ui]) over dst_ui
    scatter_edges<<<eblocks, 256, 0, stream>>>(user_feat, norm_ui, src_ui, dst_ui,
                                               item_h, n_edges);

    // Normalize both halves: 200000 rows -> 12500 tiles, 8 tiles per block.
    int n_rows  = 2 * N_NODES;
    int tiles   = (n_rows + 15) / 16;
    int nblocks = (tiles + 7) / 8;
    l2norm_rows<<<nblocks, 256, 0, stream>>>(out, n_rows);
}